// MultiLabelBCE_85023172591651
// MI455X (gfx1250) — compile-verified
//
#include <hip/hip_runtime.h>
#include <hip/hip_bf16.h>
#include <math.h>

// ---------------------------------------------------------------------------
// Problem constants (from reference): B=32768, D=512, C=527
// ---------------------------------------------------------------------------
constexpr int BB    = 32768;   // batch rows
constexpr int DD    = 512;     // feature dim (K)
constexpr int CC    = 527;     // classes
constexpr int NPAD  = 528;     // padded class dim (33 * 16 = 11 * 48)
constexpr int NBLK  = 48;      // N columns per block (3 tiles of 16 per wave)
constexpr int NGRID = NPAD / NBLK;   // 11 column blocks
constexpr int MBLK  = 128;     // rows per block (8 waves * 16)
constexpr int KCH   = 32;      // K chunk staged in LDS
constexpr int NCH   = DD / KCH;      // 16 chunks
constexpr int LDST  = 36;      // LDS row stride in floats (16B aligned, conflict-free)

constexpr int A_BUF = MBLK * LDST;       // 4608 floats per A buffer
constexpr int B_OFF = 2 * A_BUF;         // 9216: B buffers after double-buffered A
constexpr int B_BUF = NBLK * LDST;       // 1728 floats per B buffer
constexpr int LDS_FLOATS = B_OFF + 2 * B_BUF;   // 12672 floats = 50688 B

constexpr long long LOGITS_ELEMS = (long long)BB * NPAD;   // ws layout: logits first
constexpr int RED_BLOCKS = BB / 8;                         // 4096 partial pairs

typedef __attribute__((ext_vector_type(2))) float v2f;
typedef __attribute__((ext_vector_type(8))) float v8f;

// ---------------------------------------------------------------------------
// Async staging of one K chunk directly into LDS (GLOBAL_LOAD_ASYNC_TO_LDS_*).
// Exactly 7 async instructions per thread (uniform per wave): 4xb128 A, 3xb64 B.
// ---------------------------------------------------------------------------
__device__ __forceinline__ void stage_chunk_async(
    const float* __restrict__ x, const float* __restrict__ W,
    int t, int rowBase, int colBase, unsigned ldsByteBase, int buf, int k0)
{
    // ---- A: 128 rows x 32 floats ----
    const int aseg = t & 7;
    #pragma unroll
    for (int rr = 0; rr < 4; ++rr) {
        const int r = (t >> 3) + 32 * rr;
        const float* g  = &x[(size_t)(rowBase + r) * DD + k0 + aseg * 4];
        const unsigned l = ldsByteBase
                         + (unsigned)((buf * A_BUF + r * LDST + aseg * 4) * 4);
        asm volatile("global_load_async_to_lds_b128 %0, %1, off"
                     :: "v"(l), "v"(g) : "memory");
    }
    // ---- B: 48 class-rows x 32 floats (clamp pad column 527) ----
    #pragma unroll
    for (int q = 0; q < 3; ++q) {
        const int idx = t + 256 * q;
        const int nr  = idx >> 4;       // 0..47
        const int seg = idx & 15;       // floats [2*seg, 2*seg+1]
        int col = colBase + nr; col = (col < CC) ? col : (CC - 1);
        const float* g  = &W[(size_t)col * DD + k0 + seg * 2];
        const unsigned l = ldsByteBase
                         + (unsigned)((B_OFF + buf * B_BUF + nr * LDST + seg * 2) * 4);
        asm volatile("global_load_async_to_lds_b64 %0, %1, off"
                     :: "v"(l), "v"(g) : "memory");
    }
}

// ---------------------------------------------------------------------------
// Kernel 1: logits = x @ W^T + b   via V_WMMA_F32_16X16X4_F32
// grid = (11, 256), block = 256 (8 waves). Each wave: 16x48 output strip.
// Double-buffered LDS, async global->LDS staging pipelined with WMMA.
// ---------------------------------------------------------------------------
__global__ __launch_bounds__(256)
void mlbce_gemm_kernel(const float* __restrict__ x,
                       const float* __restrict__ W,
                       const float* __restrict__ bias,
                       float* __restrict__ logits)
{
    __shared__ float lds[LDS_FLOATS];

    const int t       = threadIdx.x;
    const int wave    = t >> 5;
    const int lane    = t & 31;
    const int half    = lane >> 4;        // K slice select: lanes 16-31 -> K=2,3
    const int lr      = lane & 15;        // M (A) / N (B) index within 16-tile

    const int rowBase = blockIdx.y * MBLK;          // x row base of this block
    const int colBase = blockIdx.x * NBLK;          // class column base

    const unsigned ldsByteBase = (unsigned)(uintptr_t)(&lds[0]);

    v8f acc0 = {}, acc1 = {}, acc2 = {};   // 16x16 f32 tiles (documented C/D layout)

    // Prologue: kick off chunk 0 into buffer 0
    stage_chunk_async(x, W, t, rowBase, colBase, ldsByteBase, 0, 0);

    for (int ch = 0; ch < NCH; ++ch) {
        const int cur = ch & 1;
        __syncthreads();   // all waves done reading buffer (1-cur) from chunk ch-1
        if (ch + 1 < NCH) {
            stage_chunk_async(x, W, t, rowBase, colBase, ldsByteBase,
                              1 - cur, (ch + 1) * KCH);
            asm volatile("s_wait_asynccnt 0x7" ::: "memory");  // chunk ch landed
        } else {
            asm volatile("s_wait_asynccnt 0x0" ::: "memory");
        }
        __syncthreads();   // all waves' async writes for chunk ch visible

        const float* la = &lds[cur * A_BUF + (wave * 16 + lr) * LDST];
        const float* lb = &lds[B_OFF + cur * B_BUF + lr * LDST];
        #pragma unroll
        for (int s = 0; s < 8; ++s) {
            const int kk = 4 * s + 2 * half;
            const v2f av = *(const v2f*)&la[kk];
            const v2f b0 = *(const v2f*)&lb[kk];
            const v2f b1 = *(const v2f*)&lb[16 * LDST + kk];
            const v2f b2 = *(const v2f*)&lb[32 * LDST + kk];
            acc0 = __builtin_amdgcn_wmma_f32_16x16x4_f32(
                false, av, false, b0, (short)0, acc0, false, false);
            acc1 = __builtin_amdgcn_wmma_f32_16x16x4_f32(
                false, av, false, b1, (short)0, acc1, false, false);
            acc2 = __builtin_amdgcn_wmma_f32_16x16x4_f32(
                false, av, false, b2, (short)0, acc2, false, false);
        }
    }

    // ---- epilogue: add bias, store three 16x16 tiles ----
    const int mBase = rowBase + wave * 16 + 8 * half;   // VGPR j -> M = j + 8*half
    #pragma unroll
    for (int nt = 0; nt < 3; ++nt) {
        const int col = colBase + nt * 16 + lr;
        const float bval = (col < CC) ? bias[col] : 0.0f;
        const v8f acc = (nt == 0) ? acc0 : (nt == 1) ? acc1 : acc2;
        #pragma unroll
        for (int j = 0; j < 8; ++j) {
            logits[(long long)(mBase + j) * NPAD + col] = acc[j] + bval;
        }
    }
}

// ---------------------------------------------------------------------------
// Wave (32-lane) sum reductions
// ---------------------------------------------------------------------------
__device__ __forceinline__ float waveSumF(float v) {
    #pragma unroll
    for (int off = 16; off > 0; off >>= 1) v += __shfl_xor(v, off, 32);
    return v;
}
__device__ __forceinline__ int waveSumI(int v) {
    #pragma unroll
    for (int off = 16; off > 0; off >>= 1) v += __shfl_xor(v, off, 32);
    return v;
}

__device__ __forceinline__ float softplus_stable(float t) {
    // log(1 + exp(t)) = max(t,0) + log1p(exp(-|t|))
    return fmaxf(t, 0.0f) + log1pf(expf(-fabsf(t)));
}

// ---------------------------------------------------------------------------
// Kernel 2: per-row BCE loss + top-k hit score. One wave per row.
// block = 256 (8 waves) -> 8 rows per block; grid = 4096.
// ---------------------------------------------------------------------------
__global__ __launch_bounds__(256)
void mlbce_rowstat_kernel(const float* __restrict__ logits,
                          const float* __restrict__ y,
                          const float* __restrict__ pos_weight,
                          float* __restrict__ partLoss,
                          float* __restrict__ partScore)
{
    __shared__ float wloss[8];
    __shared__ float wscore[8];

    const int wave = threadIdx.x >> 5;
    const int lane = threadIdx.x & 31;
    const int row  = blockIdx.x * 8 + wave;

    const float* zrow = &logits[(long long)row * NPAD];
    const float* yrow = &y[(long long)row * CC];

    float z[17];
    float yv[17];
    float lossLoc = 0.0f;
    int   kLoc    = 0;

    #pragma unroll
    for (int i = 0; i < 17; ++i) {
        const int c = lane + 32 * i;
        if (c < CC) {
            const float zz = zrow[c];
            const float yy = yrow[c];
            z[i]  = zz;
            yv[i] = yy;
            const float pw = pos_weight[c];
            // -(pw*y*logsig(z) + (1-y)*logsig(-z)) = pw*y*sp(-z) + (1-y)*sp(z)
            lossLoc += pw * yy * softplus_stable(-zz)
                     + (1.0f - yy) * softplus_stable(zz);
            kLoc += (yy > 0.5f) ? 1 : 0;
        } else {
            z[i]  = -INFINITY;
            yv[i] = 0.0f;
        }
    }

    const float lossRow = waveSumF(lossLoc);
    const int   k       = waveSumI(kLoc);      // >= 1 by construction

    // ---- count hits: positive c is selected iff
    //      #{j: z_j > z_c  or (z_j == z_c and j > c)} < k  (stable-argsort ties)
    float hits = 0.0f;
    #pragma unroll 1
    for (int i = 0; i < 17; ++i) {
        unsigned m = (unsigned)__ballot(yv[i] > 0.5f);
        while (m) {
            const int s = __ffs(m) - 1;
            m &= m - 1;
            const int   c  = 32 * i + s;
            const float zc = __shfl(z[i], s, 32);
            int cnt = 0;
            #pragma unroll
            for (int j = 0; j < 17; ++j) {
                const int cj = lane + 32 * j;
                const bool gt = (z[j] > zc) || ((z[j] == zc) && (cj > c));
                cnt += gt ? 1 : 0;   // OOB lanes hold -inf -> never counted
            }
            cnt = waveSumI(cnt);
            hits += (cnt < k) ? 1.0f : 0.0f;   // identical on all lanes
        }
    }
    const float scoreRow = hits / (float)k;

    if (lane == 0) { wloss[wave] = lossRow; wscore[wave] = scoreRow; }
    __syncthreads();
    if (threadIdx.x == 0) {
        float ls = 0.0f, sc = 0.0f;
        #pragma unroll
        for (int w = 0; w < 8; ++w) { ls += wloss[w]; sc += wscore[w]; }
        partLoss[blockIdx.x]  = ls;
        partScore[blockIdx.x] = sc;
    }
}

// ---------------------------------------------------------------------------
// Kernel 3: deterministic final reduction of 4096 partial pairs.
// ---------------------------------------------------------------------------
__global__ __launch_bounds__(256)
void mlbce_finalize_kernel(const float* __restrict__ partLoss,
                           const float* __restrict__ partScore,
                           float* __restrict__ out)
{
    __shared__ float sl[256];
    __shared__ float ss[256];
    const int t = threadIdx.x;
    float ls = 0.0f, sc = 0.0f;
    for (int i = t; i < RED_BLOCKS; i += 256) { ls += partLoss[i]; sc += partScore[i]; }
    sl[t] = ls; ss[t] = sc;
    __syncthreads();
    for (int off = 128; off > 0; off >>= 1) {
        if (t < off) { sl[t] += sl[t + off]; ss[t] += ss[t + off]; }
        __syncthreads();
    }
    if (t == 0) {
        out[0] = sl[0] / ((float)BB * (float)CC);   // mean BCE loss
        out[1] = ss[0] / (float)BB;                 // mean top-k score
    }
}

// ---------------------------------------------------------------------------
// Host-side launcher
// ---------------------------------------------------------------------------
extern "C" void kernel_launch(void* const* d_in, const int* in_sizes, int n_in,
                              void* d_out, int out_size, void* d_ws, size_t ws_size,
                              hipStream_t stream)
{
    const float* x  = (const float*)d_in[0];   // [B, D]
    const float* y  = (const float*)d_in[1];   // [B, C]
    const float* W  = (const float*)d_in[2];   // [C, D]
    const float* b  = (const float*)d_in[3];   // [C]
    const float* pw = (const float*)d_in[4];   // [C]

    float* ws        = (float*)d_ws;
    float* logits    = ws;                                  // [B, 528]
    float* partLoss  = ws + LOGITS_ELEMS;                   // [4096]
    float* partScore = ws + LOGITS_ELEMS + RED_BLOCKS;      // [4096]
    float* out       = (float*)d_out;                       // [2]: loss, score

    dim3 gridG(NGRID, BB / MBLK);   // (11, 256)
    mlbce_gemm_kernel<<<gridG, 256, 0, stream>>>(x, W, b, logits);

    mlbce_rowstat_kernel<<<RED_BLOCKS, 256, 0, stream>>>(logits, y, pw,
                                                         partLoss, partScore);

    mlbce_finalize_kernel<<<1, 256, 0, stream>>>(partLoss, partScore, out);
}